// DynamicHead_90907277787466
// MI455X (gfx1250) — compile-verified
//
#include <hip/hip_runtime.h>

typedef __attribute__((ext_vector_type(2))) float v2f;
typedef __attribute__((ext_vector_type(8))) float v8f;

#define OBJ     50
#define CPG     8
#define HW      (128 * 128)
#define PSPLIT  8                         // pixel chunks per (b,g) pair
#define TILES   (HW / 16)                 // 1024 16-pixel tiles per (b,g)
#define TPC     (TILES / PSPLIT)          // 128 tiles per chunk
#define WAVES   8                         // 256-thread block = 8 wave32

// out[b,n,p] = sum_c kernel[b,n,c] * f[b, n*CPG+c, p]
// One wave: A = group weights (rows replicated), B = 4ch x 16px of f,
// two chained V_WMMA_F32_16X16X4_F32 give the exact fp32 8-channel dot
// for 16 pixels; c[0] lanes 0-15 = the 16 outputs.
__global__ __launch_bounds__(256) void DynamicHead_wmma_f32(
    const float* __restrict__ f,      // [8, 400, 128, 128]
    const float* __restrict__ kern,   // [8, 50, 8, 1, 1]
    float* __restrict__ out)          // [8, 50, 128, 128]
{
    const int blk   = blockIdx.x;
    const int chunk = blk & (PSPLIT - 1);
    const int bg    = blk >> 3;                    // (b*50 + g), 0..399
    const int lane  = threadIdx.x & 31;
    const int wave  = threadIdx.x >> 5;
    const int hi    = lane >> 4;                   // 0: lanes 0-15, 1: lanes 16-31
    const int ln    = lane & 15;

    // base of this group's 8 input channels: (b*400 + g*8)*HW == bg*8*HW
    const float* fp = f    + (size_t)bg * CPG * HW;
    const float* kw = kern + (size_t)bg * CPG;
    float*       op = out  + (size_t)bg * HW;

    // A matrix (16x4 fp32): VGPR0 = K=0 (lanes 0-15) / K=2 (lanes 16-31), VGPR1 = K=1/K=3.
    // All 16 M-rows identical (weight broadcast).
    v2f a01, a23;
    a01.x = kw[0 + 2 * hi];
    a01.y = kw[1 + 2 * hi];
    a23.x = kw[4 + 2 * hi];
    a23.y = kw[5 + 2 * hi];

    // Per-lane element offsets for the B-matrix loads (channel-pair striping).
    const int off0 = (0 + 2 * hi) * HW + ln;       // ch 0 / ch 2
    const int off1 = (1 + 2 * hi) * HW + ln;       // ch 1 / ch 3
    const int off2 = (4 + 2 * hi) * HW + ln;       // ch 4 / ch 6
    const int off3 = (5 + 2 * hi) * HW + ln;       // ch 5 / ch 7

#pragma unroll 4
    for (int i = 0; i < TPC / WAVES; ++i) {
        const int pbase = (chunk * TPC + wave + i * WAVES) * 16;

        // B matrix (4x16 fp32): row K striped across lanes within one VGPR.
        v2f b01, b23;
        b01.x = fp[off0 + pbase];
        b01.y = fp[off1 + pbase];
        b23.x = fp[off2 + pbase];
        b23.y = fp[off3 + pbase];

        v8f c = {};
        // D = A(ch0..3) x B + 0, then += A(ch4..7) x B  -> exact fp32 K=8 reduction
        c = __builtin_amdgcn_wmma_f32_16x16x4_f32(false, a01, false, b01,
                                                  (short)0, c, false, false);
        c = __builtin_amdgcn_wmma_f32_16x16x4_f32(false, a23, false, b23,
                                                  (short)0, c, false, false);

        // c[0] lanes 0-15 = out[g, pbase+ln]; lanes 16-31 are replicas -> half-wave store.
        if (hi == 0) {
            op[pbase + ln] = c[0];
        }
    }
}

extern "C" void kernel_launch(void* const* d_in, const int* in_sizes, int n_in,
                              void* d_out, int out_size, void* d_ws, size_t ws_size,
                              hipStream_t stream) {
    (void)in_sizes; (void)n_in; (void)out_size; (void)d_ws; (void)ws_size;
    const float* f    = (const float*)d_in[0];   // [8,400,128,128] fp32
    const float* kern = (const float*)d_in[1];   // [8,50,8,1,1]    fp32
    float*       out  = (float*)d_out;           // [8,50,128,128]  fp32

    const int n_bg   = 8 * OBJ;                  // 400 (b,g) pairs
    dim3 grid(n_bg * PSPLIT);                    // 3200 blocks
    dim3 block(256);                             // 8 wave32 per block
    DynamicHead_wmma_f32<<<grid, block, 0, stream>>>(f, kern, out);
}